// SamplerQNN_65481071403411
// MI455X (gfx1250) — compile-verified
//
#include <hip/hip_runtime.h>
#include <math.h>

#define N_QUBITS 12
#define DIM 4096
#define BATCH 8192

typedef __attribute__((ext_vector_type(2))) float v2f;
typedef __attribute__((ext_vector_type(8))) float v8f;

// ---------------------------------------------------------------------------
// Kernel 1: compute rows 0 and 1 of the fixed variational unitary V by
// simulating V^dagger |0> and V^dagger |1> in LDS (one workgroup), then store
// A[256][64] = [Re w0 ; -Im w0 ; Re w1 ; -Im w1] reshaped (u,v) into d_ws.
// Qubit i lives at bit position (11 - i) of the state index.
// ---------------------------------------------------------------------------
__global__ __launch_bounds__(256) void qnn_build_rows(const float* __restrict__ params,
                                                      float* __restrict__ A) {
  __shared__ float sr[2 * DIM];  // 32 KB
  __shared__ float si[2 * DIM];  // 32 KB
  const int tid = threadIdx.x;

  for (int j = tid; j < 2 * DIM; j += 256) { sr[j] = 0.f; si[j] = 0.f; }
  __syncthreads();
  if (tid == 0) { sr[0] = 1.f; sr[DIM + 1] = 1.f; }  // |0> and |1>
  __syncthreads();

  // Apply a 1-qubit gate (complex 2x2) at bit position `pos` to both states.
  auto apply1q = [&](int pos, float u00r, float u00i, float u01r, float u01i,
                     float u10r, float u10i, float u11r, float u11i) {
    for (int t = tid; t < 2 * (DIM / 2); t += 256) {
      int st = t >> 11;
      int tt = t & (DIM / 2 - 1);
      int j0 = ((tt >> pos) << (pos + 1)) | (tt & ((1 << pos) - 1));
      int j1 = j0 | (1 << pos);
      int b = st * DIM;
      float a0r = sr[b + j0], a0i = si[b + j0];
      float a1r = sr[b + j1], a1i = si[b + j1];
      float n0r = u00r * a0r - u00i * a0i + u01r * a1r - u01i * a1i;
      float n0i = u00r * a0i + u00i * a0r + u01r * a1i + u01i * a1r;
      float n1r = u10r * a0r - u10i * a0i + u11r * a1r - u11i * a1i;
      float n1i = u10r * a0i + u10i * a0r + u11r * a1i + u11i * a1r;
      sr[b + j0] = n0r; si[b + j0] = n0i;
      sr[b + j1] = n1r; si[b + j1] = n1i;
    }
    __syncthreads();
  };

  // CNOT (self-inverse): control wire c -> bit pc = 11-c, target bit pc-1.
  auto cnot = [&](int ctrl) {
    int pc = 11 - ctrl;
    int pt = pc - 1;
    for (int t = tid; t < 2 * (DIM / 4); t += 256) {
      int st = t >> 10;
      int tt = t & (DIM / 4 - 1);
      int low = tt & ((1 << pt) - 1);
      int high = tt >> pt;
      int j0 = (high << (pt + 2)) | (1 << pc) | low;  // ctrl=1, tgt=0
      int j1 = j0 | (1 << pt);                        // ctrl=1, tgt=1
      int b = st * DIM;
      float r0 = sr[b + j0], i0 = si[b + j0];
      sr[b + j0] = sr[b + j1]; si[b + j0] = si[b + j1];
      sr[b + j1] = r0;         si[b + j1] = i0;
    }
    __syncthreads();
  };

  // V = Prod_layers [ rotations(RY,RX,RZ per qubit) then CNOT ladder ].
  // Apply V^dagger: reverse layers; per layer undo CNOTs (reverse order),
  // then per qubit apply RZ(-t2), RX(-t1), RY(-t0).
  for (int layer = 2; layer >= 0; --layer) {
    for (int q = 10; q >= 0; --q) cnot(q);
    for (int q = 11; q >= 0; --q) {
      const float* pp = params + (layer * 12 + q) * 3;
      float t0 = pp[0], t1 = pp[1], t2 = pp[2];
      int pos = 11 - q;
      {  // RZ(-t2): diag(cos h - i sin h, cos h + i sin h), h = -t2/2
        float h = -0.5f * t2; float c = cosf(h), s = sinf(h);
        apply1q(pos, c, -s, 0.f, 0.f, 0.f, 0.f, c, s);
      }
      {  // RX(-t1): [[c, -i s], [-i s, c]] with h = -t1/2
        float h = -0.5f * t1; float c = cosf(h), s = sinf(h);
        apply1q(pos, c, 0.f, 0.f, -s, 0.f, -s, c, 0.f);
      }
      {  // RY(-t0): [[c, -s], [s, c]] with h = -t0/2
        float h = -0.5f * t0; float c = cosf(h), s = sinf(h);
        apply1q(pos, c, 0.f, -s, 0.f, s, 0.f, c, 0.f);
      }
    }
  }

  // Emit A[r][v], r = block*64 + u, j = u*64 + v.
  // block 0: Re w0, 1: -Im w0, 2: Re w1, 3: -Im w1  (conj of row k of V).
  {
    int r = tid;
    int block = r >> 6, u = r & 63;
    int k = block >> 1;
    bool re = (block & 1) == 0;
    for (int v = 0; v < 64; ++v) {
      int j = (u << 6) | v;
      A[r * 64 + v] = re ? sr[k * DIM + j] : -si[k * DIM + j];
    }
  }
}

// ---------------------------------------------------------------------------
// Kernel 2: per workgroup (256 thr = 8 waves) cache A (64 KB) in LDS; each
// wave handles 16 samples. B operand (q-factors, K=64 contraction dim) and
// p-factors are built in registers from 12 trig values per sample. 16 M-tiles
// x 16 V_WMMA_F32_16X16X4_F32 each, then a p-weighted reduction of the C tile
// with a cross-half __shfl_xor(16) under wave32.
// Trig args are x/2 in [-pi/2, pi/2], so the hardware v_sin/v_cos path
// (__sinf/__cosf) is accurate and avoids libm range reduction in the hot loop.
// ---------------------------------------------------------------------------
__global__ __launch_bounds__(256) void qnn_contract(const float* __restrict__ x,
                                                    const float* __restrict__ A,
                                                    float* __restrict__ out) {
  __shared__ float Ash[256 * 64];  // 64 KB
  const int tid = threadIdx.x;
  for (int i = tid; i < 256 * 64; i += 256) Ash[i] = A[i];
  __syncthreads();

  const int wave = tid >> 5;
  const int lane = tid & 31;
  const int half = lane >> 4;   // which 16-lane half
  const int n = lane & 15;      // sample column within the wave tile
  const int s = blockIdx.x * 128 + wave * 16 + n;

  const float* xs = x + s * 12;
  float cs[12], sn[12];
#pragma unroll
  for (int i = 0; i < 12; ++i) {
    float h = 0.5f * xs[i];     // bounded: |h| <= pi/2
    cs[i] = __cosf(h);
    sn[i] = __sinf(h);
  }

  // B operand registers: lane supplies B[K=4k+c0][n], B[4k+c0+1][n];
  // q[v] = prod over qubits 6..11 (v bit (5-i) selects sin/cos of qubit 6+i).
  const int c0 = half * 2;
  v2f bq[16];
#pragma unroll
  for (int k = 0; k < 16; ++k) {
    int v0 = 4 * k + c0;
    float q0 = 1.f, q1 = 1.f;
#pragma unroll
    for (int i = 0; i < 6; ++i) {
      q0 *= ((v0 >> (5 - i)) & 1) ? sn[6 + i] : cs[6 + i];
      q1 *= (((v0 + 1) >> (5 - i)) & 1) ? sn[6 + i] : cs[6 + i];
    }
    bq[k].x = q0;
    bq[k].y = q1;
  }

  // p-factors for the C-tile reduction: u = m4*16 + half*8 + g.
  float pv[32];
#pragma unroll
  for (int m4 = 0; m4 < 4; ++m4) {
#pragma unroll
    for (int g = 0; g < 8; ++g) {
      int u = m4 * 16 + half * 8 + g;
      float p = 1.f;
#pragma unroll
      for (int i = 0; i < 6; ++i) p *= ((u >> (5 - i)) & 1) ? sn[i] : cs[i];
      pv[m4 * 8 + g] = p;
    }
  }

  float acc0 = 0.f, acc1 = 0.f, acc2 = 0.f, acc3 = 0.f;

#pragma unroll
  for (int mt = 0; mt < 16; ++mt) {
    v8f c = {};
    const float* arow = &Ash[(mt * 16 + n) * 64 + c0];
#pragma unroll
    for (int k = 0; k < 16; ++k) {
      v2f a = *(const v2f*)(arow + 4 * k);
      c = __builtin_amdgcn_wmma_f32_16x16x4_f32(false, a, false, bq[k], (short)0, c,
                                                false, false);
    }
    float r = 0.f;
#pragma unroll
    for (int g = 0; g < 8; ++g) r += pv[(mt & 3) * 8 + g] * c[g];
    r += __shfl_xor(r, 16, 32);  // combine M 0-7 and 8-15 halves (wave32)
    if ((mt >> 2) == 0) acc0 += r;
    else if ((mt >> 2) == 1) acc1 += r;
    else if ((mt >> 2) == 2) acc2 += r;
    else acc3 += r;
  }

  if (lane < 16) {
    float p20 = acc0 * acc0 + acc1 * acc1;  // |a0|^2
    float p21 = acc2 * acc2 + acc3 * acc3;  // |a1|^2
    float inv = 1.f / (p20 + p21);
    out[2 * s + 0] = p20 * inv;
    out[2 * s + 1] = p21 * inv;
  }
}

extern "C" void kernel_launch(void* const* d_in, const int* in_sizes, int n_in,
                              void* d_out, int out_size, void* d_ws, size_t ws_size,
                              hipStream_t stream) {
  const float* x = (const float*)d_in[0];       // (8192, 12) f32
  const float* params = (const float*)d_in[1];  // (3, 12, 3) f32
  float* out = (float*)d_out;                   // (8192, 2) f32
  float* A = (float*)d_ws;                      // 256*64 f32 = 64 KB scratch

  qnn_build_rows<<<1, 256, 0, stream>>>(params, A);
  qnn_contract<<<BATCH / 128, 256, 0, stream>>>(x, A, out);
}